// One2ManyBezierPointsHungarianMatcher_28295244546870
// MI455X (gfx1250) — compile-verified
//
#include <hip/hip_runtime.h>
#include <hip/hip_bf16.h>

typedef __attribute__((ext_vector_type(2))) float v2f;
typedef __attribute__((ext_vector_type(8))) float v8f;

#define BS      4
#define NQ      1000
#define NGT     32
#define SPTS    100
#define ROWF    200              // floats per sampled row (100 pts * 2)
#define KMATCH  13
#define NMATCH  (NGT * KMATCH)   // 416
#define CTILES  2                // column tiles per wave

// ---------------------------------------------------------------------------
// Bernstein basis for cubic Bezier sampled at s/99, s in [0,100)
// ---------------------------------------------------------------------------
__device__ __forceinline__ float bern(int s, int k) {
  float t = (float)s * (1.0f / 99.0f);
  float u = 1.0f - t;
  switch (k) {
    case 0:  return u * u * u;
    case 1:  return 3.0f * t * u * u;
    case 2:  return 3.0f * t * t * u;
    default: return t * t * t;
  }
}

// ---------------------------------------------------------------------------
// Bezier sampling as GEMM: D[100 x 2N] = B[100 x 4] * C[4 x 2N]
// via V_WMMA_F32_16X16X4_F32 (K=4 native).  One wave handles CTILES adjacent
// 16x16 tiles sharing one A operand (Bernstein tile built in registers).
//   blockIdx.x = column super-tile (cols = 2*ncurves, multiple of 16*CTILES)
//   blockIdx.y = row tile of sample points (7 tiles cover 100)
// ctrl: [ncurves][4][2] row-major, pts out: [ncurves][200] (x0,y0,x1,y1,...)
// ---------------------------------------------------------------------------
__global__ void bezier_sample_wmma(const float* __restrict__ ctrl,
                                   float* __restrict__ pts) {
  const int lane = threadIdx.x;        // 0..31, wave32
  const int half = lane >> 4;          // 0: K={0,1}, 1: K={2,3}
  const int kb   = half * 2;

  // A operand: 16x4 f32 tile of the Bernstein matrix.
  // Layout: lanes 0-15 row M=lane (VGPR0=K0, VGPR1=K1); lanes 16-31 K2/K3.
  const int m = blockIdx.y * 16 + (lane & 15);
  v2f a;
  if (m < SPTS) { a.x = bern(m, kb); a.y = bern(m, kb + 1); }
  else          { a.x = 0.0f;        a.y = 0.0f; }

  // Scalar-uniform: is this row tile entirely in range?
  const bool full_tile = (blockIdx.y * 16 + 15) < SPTS;

#pragma unroll
  for (int tt = 0; tt < CTILES; ++tt) {
    // B operand: 4x16 f32 tile of control points; col c -> curve c>>1, dim c&1
    const int col = (blockIdx.x * CTILES + tt) * 16 + (lane & 15);
    const int n = col >> 1, d = col & 1;
    const float* cp = ctrl + n * 8 + d;  // ctrl[n][k][d] = ctrl[n*8 + k*2 + d]
    v2f b;
    b.x = cp[kb * 2];
    b.y = cp[(kb + 1) * 2];

    v8f acc = {};
    // (neg_a, A, neg_b, B, c_mod, C, reuse_a, reuse_b)
    acc = __builtin_amdgcn_wmma_f32_16x16x4_f32(
        false, a, false, b, (short)0, acc, false, false);

    // D layout: VGPR v holds M=v (lanes 0-15) / M=v+8 (lanes 16-31), N=lane&15
    float* prow = pts + n * ROWF + d;
    const int s0 = blockIdx.y * 16 + half * 8;
    if (full_tile) {
#pragma unroll
      for (int v = 0; v < 8; ++v)
        prow[2 * (s0 + v)] = acc[v];
    } else {
#pragma unroll
      for (int v = 0; v < 8; ++v) {
        const int s = s0 + v;
        if (s < SPTS) prow[2 * s] = acc[v];
      }
    }
  }
}

// ---------------------------------------------------------------------------
// Focal-loss class cost per query (nc == 1, labels all zero by construction)
// ---------------------------------------------------------------------------
__global__ void focal_class_cost(const float* __restrict__ logits,
                                 float* __restrict__ cls, int n) {
  const int i = blockIdx.x * blockDim.x + threadIdx.x;
  if (i >= n) return;
  const float p   = 1.0f / (1.0f + __expf(-logits[i]));
  const float pos = 0.25f * (1.0f - p) * (1.0f - p) * (-__logf(p + 1e-8f));
  const float neg = 0.75f * p * p * (-log1pf(-p + 1e-8f));
  cls[i] = pos - neg;
}

// ---------------------------------------------------------------------------
// C[b][j][q] = 2*cls + 5*L1(out_pts[b*NQ+q], tgt_pts[b*NGT+j])
// One block per (b, j); target row staged in LDS.
// ---------------------------------------------------------------------------
__global__ void pair_cost(const float* __restrict__ out_pts,
                          const float* __restrict__ tgt_pts,
                          const float* __restrict__ cls,
                          float* __restrict__ Cmat) {
  __shared__ __align__(16) float trow[ROWF];
  const int b = blockIdx.x >> 5;
  const int j = blockIdx.x & 31;
  for (int t = threadIdx.x; t < ROWF; t += blockDim.x)
    trow[t] = tgt_pts[(b * NGT + j) * ROWF + t];
  __syncthreads();

  for (int q = threadIdx.x; q < NQ; q += blockDim.x) {
    const float4* pr = (const float4*)(out_pts + (size_t)(b * NQ + q) * ROWF);
    if (q + (int)blockDim.x < NQ)
      __builtin_prefetch(out_pts + (size_t)(b * NQ + q + blockDim.x) * ROWF, 0, 1);
    float sum = 0.0f;
#pragma unroll 4
    for (int t = 0; t < ROWF / 4; ++t) {
      const float4 v = pr[t];
      const float4 w = ((const float4*)trow)[t];
      sum += fabsf(v.x - w.x) + fabsf(v.y - w.y) +
             fabsf(v.z - w.z) + fabsf(v.w - w.w);
    }
    Cmat[(b * NGT + j) * NQ + q] = 2.0f * cls[b * NQ + q] + 5.0f * sum;
  }
}

// ---------------------------------------------------------------------------
// Per-image assignment: top-13 per column (stable), argmin over selected
// columns via 64-bit LDS atomicMin (key = ordered-float-bits<<32 | j),
// then sorted-unique-padded src list.  One block per image.
// ---------------------------------------------------------------------------
__global__ void assign_kernel(const float* __restrict__ Cmat,
                              float* __restrict__ out) {
  __shared__ unsigned long long best[NQ];
  __shared__ int flag[NQ];
  __shared__ int uniq[NMATCH];
  __shared__ int cnt;
  const int b = blockIdx.x;

  for (int q = threadIdx.x; q < NQ; q += blockDim.x) {
    best[q] = 0xFFFFFFFFFFFFFFFFull;
    flag[q] = 0;
  }
  __syncthreads();

  if (threadIdx.x < NGT) {
    const int j = threadIdx.x;
    const float* colp = Cmat + (b * NGT + j) * NQ;
    float c13[KMATCH];
    int   i13[KMATCH];
#pragma unroll
    for (int r = 0; r < KMATCH; ++r) { c13[r] = 3.0e38f; i13[r] = 0; }
    for (int q = 0; q < NQ; ++q) {
      const float c = colp[q];
      if (c < c13[KMATCH - 1]) {              // strict <: earlier index wins ties
        int p = KMATCH - 1;
        while (p > 0 && c < c13[p - 1]) {
          c13[p] = c13[p - 1]; i13[p] = i13[p - 1]; --p;
        }
        c13[p] = c; i13[p] = q;
      }
    }
#pragma unroll
    for (int r = 0; r < KMATCH; ++r) {
      const int q = i13[r];
      flag[q] = 1;
      unsigned u = __float_as_uint(c13[r]);   // order-preserving float -> uint
      u = (u & 0x80000000u) ? ~u : (u | 0x80000000u);
      const unsigned long long key =
          (((unsigned long long)u) << 32) | (unsigned long long)j;
      atomicMin(&best[q], key);               // ds_min_u64
    }
  }
  __syncthreads();

  if (threadIdx.x == 0) {
    int c = 0;
    for (int q = 0; q < NQ; ++q)
      if (flag[q]) uniq[c++] = q;             // ascending => sorted unique
    cnt = c;
    for (int i = c; i < NMATCH; ++i) uniq[i] = NQ;  // sentinel pad = nq
  }
  __syncthreads();

  if (threadIdx.x < NMATCH) {
    const int q = uniq[threadIdx.x];
    out[b * NMATCH + threadIdx.x] = (float)q;
    float tg = -1.0f;
    if (q < NQ) tg = (float)(int)(best[q] & 0xFFFFFFFFull);
    out[BS * NMATCH + b * NMATCH + threadIdx.x] = tg;
  }
  if (threadIdx.x == 0) out[2 * BS * NMATCH + b] = (float)cnt;
}

// ---------------------------------------------------------------------------
extern "C" void kernel_launch(void* const* d_in, const int* in_sizes, int n_in,
                              void* d_out, int out_size, void* d_ws, size_t ws_size,
                              hipStream_t stream) {
  (void)in_sizes; (void)n_in; (void)out_size; (void)ws_size;

  const float* pred_logits  = (const float*)d_in[0];  // [4,1000,1]
  const float* pred_beziers = (const float*)d_in[1];  // [4,1000,8]
  // d_in[2] = tgt_labels: nc==1 so labels are identically zero; unused.
  const float* tgt_beziers  = (const float*)d_in[3];  // [4,32,4,2]

  float* ws      = (float*)d_ws;
  float* out_pts = ws;                                // 4000 * 200
  float* tgt_pts = out_pts + BS * NQ * ROWF;          //  128 * 200
  float* cls     = tgt_pts + BS * NGT * ROWF;         // 4000
  float* Cmat    = cls + BS * NQ;                     //  128 * 1000

  // Pred curves: cols = 8000 -> 250 super-tiles of 2x16; rows -> 7 tiles.
  bezier_sample_wmma<<<dim3((2 * BS * NQ) / (16 * CTILES), 7), 32, 0, stream>>>(
      pred_beziers, out_pts);
  // Tgt curves: cols = 256 -> 8 super-tiles.
  bezier_sample_wmma<<<dim3((2 * BS * NGT) / (16 * CTILES), 7), 32, 0, stream>>>(
      tgt_beziers, tgt_pts);

  focal_class_cost<<<(BS * NQ + 255) / 256, 256, 0, stream>>>(
      pred_logits, cls, BS * NQ);

  pair_cost<<<BS * NGT, 256, 0, stream>>>(out_pts, tgt_pts, cls, Cmat);

  assign_kernel<<<BS, 512, 0, stream>>>(Cmat, (float*)d_out);
}